// PointNetFeaturePropagation_5609227289179
// MI455X (gfx1250) — compile-verified
//
#include <hip/hip_runtime.h>
#include <math.h>

typedef float f2  __attribute__((ext_vector_type(2)));
typedef float f4  __attribute__((ext_vector_type(4)));
typedef float v8f __attribute__((ext_vector_type(8)));

// Problem dims (from reference setup_inputs)
constexpr int BB   = 4;
constexpr int NN   = 16384;
constexpr int SS   = 4096;
constexpr int D1   = 128;
constexpr int D2   = 256;
constexpr int CIN  = D1 + D2;   // 384
constexpr int O1   = 256;
constexpr int O2   = 128;
constexpr int MTOT = BB * NN;   // 65536
constexpr float EPS_BN = 1e-5f;

// ------------------------------------------------------------------
// 1) 3-NN search: per query point scan S candidates (LDS-tiled),
//    branchless top-3, emit indices + normalized inverse-dist weights.
// ------------------------------------------------------------------
__global__ void knn_kernel(const float* __restrict__ xyz1,
                           const float* __restrict__ xyz2,
                           int* __restrict__ outIdx,
                           float* __restrict__ outW) {
  __shared__ float lds[3 * 1024];
  const int b = blockIdx.y;
  const int q = blockIdx.x * blockDim.x + threadIdx.x;
  const float* xq = xyz1 + ((size_t)b * NN + q) * 3;
  const float qx = xq[0], qy = xq[1], qz = xq[2];

  float c0 = 3.4e38f, c1 = 3.4e38f, c2 = 3.4e38f;
  int   i0 = 0, i1 = 0, i2 = 0;

  for (int base = 0; base < SS; base += 1024) {
    const float* src = xyz2 + ((size_t)b * SS + base) * 3;
    for (int t = threadIdx.x; t < 3 * 1024; t += blockDim.x) lds[t] = src[t];
    __syncthreads();
#pragma unroll 4
    for (int j = 0; j < 1024; ++j) {
      const float dx = qx - lds[j * 3 + 0];
      const float dy = qy - lds[j * 3 + 1];
      const float dz = qz - lds[j * 3 + 2];
      const float d  = dx * dx + dy * dy + dz * dz;
      const int   s  = base + j;
      const bool lt0 = d < c0, lt1 = d < c1, lt2 = d < c2;
      c2 = lt1 ? c1 : (lt2 ? d : c2);  i2 = lt1 ? i1 : (lt2 ? s : i2);
      c1 = lt0 ? c0 : (lt1 ? d : c1);  i1 = lt0 ? i0 : (lt1 ? s : i1);
      c0 = lt0 ? d  : c0;              i0 = lt0 ? s : i0;
    }
    __syncthreads();
  }
  const float r0 = 1.f / (c0 + 1e-8f);
  const float r1 = 1.f / (c1 + 1e-8f);
  const float r2 = 1.f / (c2 + 1e-8f);
  const float inv = 1.f / (r0 + r1 + r2);
  const size_t o = ((size_t)b * NN + q) * 3;
  outIdx[o + 0] = i0; outIdx[o + 1] = i1; outIdx[o + 2] = i2;
  outW[o + 0] = r0 * inv; outW[o + 1] = r1 * inv; outW[o + 2] = r2 * inv;
}

// ------------------------------------------------------------------
// 2) Build X = concat(points1, interpolate(points2)) : [MTOT, 384]
// ------------------------------------------------------------------
__global__ void interp_concat_kernel(const float* __restrict__ points1,
                                     const float* __restrict__ points2,
                                     const int* __restrict__ idx,
                                     const float* __restrict__ w,
                                     float* __restrict__ X) {
  const int Q4 = CIN / 4;  // 96 float4 per row
  size_t t = (size_t)blockIdx.x * blockDim.x + threadIdx.x;
  size_t p = t / Q4;
  int   c4 = (int)(t % Q4);
  if (p >= (size_t)MTOT) return;
  const int b = (int)(p / NN);
  f4 v;
  if (c4 < D1 / 4) {
    v = *(const f4*)(points1 + p * (size_t)D1 + c4 * 4);
  } else {
    const int cc = c4 * 4 - D1;
    const size_t o = p * 3;
    const int j0 = idx[o + 0], j1 = idx[o + 1], j2 = idx[o + 2];
    const float w0 = w[o + 0], w1 = w[o + 1], w2 = w[o + 2];
    const float* pb = points2 + (size_t)b * SS * D2;
    f4 a0 = *(const f4*)(pb + (size_t)j0 * D2 + cc);
    f4 a1 = *(const f4*)(pb + (size_t)j1 * D2 + cc);
    f4 a2 = *(const f4*)(pb + (size_t)j2 * D2 + cc);
    v = a0 * w0 + a1 * w1 + a2 * w2;
  }
  *(f4*)(X + p * (size_t)CIN + c4 * 4) = v;
}

// ------------------------------------------------------------------
// 3) Zero the stats scratch region
// ------------------------------------------------------------------
__global__ void zero_stats_kernel(float* __restrict__ st, int n) {
  const int t = blockIdx.x * blockDim.x + threadIdx.x;
  if (t < n) st[t] = 0.f;
}

// ------------------------------------------------------------------
// 4) WMMA fp32 GEMM: out = A[M,K] * W[N,K]^T + bias, with fused
//    BN-stat accumulation (sum, sumsq of pre-BN outputs) and optional
//    fused BN+ReLU transform on the *input* A (for layer 2).
//    Block tile 128x32, BK=32, 8 waves, each wave owns 32x16 output.
// ------------------------------------------------------------------
template <int KDIM>
__global__ void gemm_wmma_kernel(const float* __restrict__ A,
                                 const float* __restrict__ W,
                                 const float* __restrict__ bias,
                                 const float* __restrict__ inScale,
                                 const float* __restrict__ inShift,
                                 float* __restrict__ out,
                                 float* __restrict__ statSum,
                                 float* __restrict__ statSq,
                                 int Ncols) {
  constexpr int BM = 128, BN = 32, BK = 32, LDT = BK + 4;  // pad: conflict-free, 16B aligned rows
  __shared__ float As[BM * LDT];
  __shared__ float Bs[BN * LDT];
  __shared__ float sSum[BN], sSq[BN];

  const int tid = threadIdx.x;
  if (tid < BN) { sSum[tid] = 0.f; sSq[tid] = 0.f; }

  const int m0 = blockIdx.x * BM;
  const int n0 = blockIdx.y * BN;
  const int wave = tid >> 5, lane = tid & 31;
  const int wm = wave & 3, wn = wave >> 2;     // 4 waves along M, 2 along N
  const int half = lane >> 4, lrow = lane & 15;

  v8f acc0 = {0.f, 0.f, 0.f, 0.f, 0.f, 0.f, 0.f, 0.f};
  v8f acc1 = {0.f, 0.f, 0.f, 0.f, 0.f, 0.f, 0.f, 0.f};

  const int lr  = tid >> 3;        // 0..31 loader row
  const int lc4 = (tid & 7) * 4;   // 0..28 loader col (floats)

  for (int kb = 0; kb < KDIM; kb += BK) {
    // --- load A tile (128 x 32), optional fused BN1+ReLU on input ---
    f4 scale, shift;
    if (inScale) {
      scale = *(const f4*)(inScale + kb + lc4);
      shift = *(const f4*)(inShift + kb + lc4);
    }
#pragma unroll
    for (int i = 0; i < 4; ++i) {
      const int row = lr + i * 32;
      f4 v = *(const f4*)(A + (size_t)(m0 + row) * KDIM + kb + lc4);
      if (inScale) {
        v = v * scale + shift;
        v.x = fmaxf(v.x, 0.f); v.y = fmaxf(v.y, 0.f);
        v.z = fmaxf(v.z, 0.f); v.w = fmaxf(v.w, 0.f);
      }
      *(f4*)(As + row * LDT + lc4) = v;
    }
    // --- load B tile (32 x 32) from W[N,K] row-major ---
    {
      f4 wv = *(const f4*)(W + (size_t)(n0 + lr) * KDIM + kb + lc4);
      *(f4*)(Bs + lr * LDT + lc4) = wv;
    }
    __syncthreads();

    // --- MACs: K steps of 4 via V_WMMA_F32_16X16X4_F32 ---
#pragma unroll
    for (int k = 0; k < BK; k += 4) {
      f2 bv = *(const f2*)(Bs + (wn * 16 + lrow) * LDT + k + 2 * half);
      f2 a0 = *(const f2*)(As + (wm * 32      + lrow) * LDT + k + 2 * half);
      f2 a1 = *(const f2*)(As + (wm * 32 + 16 + lrow) * LDT + k + 2 * half);
      acc0 = __builtin_amdgcn_wmma_f32_16x16x4_f32(false, a0, false, bv,
                                                   (short)0, acc0, false, false);
      acc1 = __builtin_amdgcn_wmma_f32_16x16x4_f32(false, a1, false, bv,
                                                   (short)0, acc1, false, false);
    }
    __syncthreads();
  }

  // --- epilogue: bias, store raw, per-channel partial stats ---
  const int col = n0 + wn * 16 + lrow;
  const float bval = bias[col];
  float psum = 0.f, psq = 0.f;
#pragma unroll
  for (int r = 0; r < 8; ++r) {
    const float y0 = acc0[r] + bval;
    const float y1 = acc1[r] + bval;
    const int row0 = m0 + wm * 32 + half * 8 + r;
    const int row1 = row0 + 16;
    out[(size_t)row0 * Ncols + col] = y0;
    out[(size_t)row1 * Ncols + col] = y1;
    psum += y0 + y1;
    psq  += y0 * y0 + y1 * y1;
  }
  atomicAdd(&sSum[wn * 16 + lrow], psum);
  atomicAdd(&sSq [wn * 16 + lrow], psq);
  __syncthreads();
  if (tid < BN) {
    atomicAdd(&statSum[n0 + tid], sSum[tid]);
    atomicAdd(&statSq [n0 + tid], sSq[tid]);
  }
}

// ------------------------------------------------------------------
// 5) Finalize BN: scale = g*rsqrt(var+eps), shift = beta - mean*scale
// ------------------------------------------------------------------
__global__ void finalize_stats_kernel(const float* __restrict__ sum,
                                      const float* __restrict__ sq,
                                      const float* __restrict__ gamma,
                                      const float* __restrict__ beta,
                                      float* __restrict__ scale,
                                      float* __restrict__ shift, int nch) {
  const int c = blockIdx.x * blockDim.x + threadIdx.x;
  if (c >= nch) return;
  const float invM = 1.f / (float)MTOT;
  const float mean = sum[c] * invM;
  const float var  = sq[c] * invM - mean * mean;
  const float s    = gamma[c] * rsqrtf(var + EPS_BN);
  scale[c] = s;
  shift[c] = beta[c] - mean * s;
}

// ------------------------------------------------------------------
// 6) Final BN2 + ReLU, in-place on d_out [MTOT, 128]
// ------------------------------------------------------------------
__global__ void bnrelu_out_kernel(float* __restrict__ out,
                                  const float* __restrict__ scale,
                                  const float* __restrict__ shift) {
  const size_t t = (size_t)blockIdx.x * blockDim.x + threadIdx.x;  // float4 index
  const int c4 = (int)(t % (O2 / 4)) * 4;
  f4 v  = *(f4*)(out + t * 4);
  f4 sc = *(const f4*)(scale + c4);
  f4 sh = *(const f4*)(shift + c4);
  v = v * sc + sh;
  v.x = fmaxf(v.x, 0.f); v.y = fmaxf(v.y, 0.f);
  v.z = fmaxf(v.z, 0.f); v.w = fmaxf(v.w, 0.f);
  *(f4*)(out + t * 4) = v;
}

// ------------------------------------------------------------------
extern "C" void kernel_launch(void* const* d_in, const int* in_sizes, int n_in,
                              void* d_out, int out_size, void* d_ws, size_t ws_size,
                              hipStream_t stream) {
  const float* xyz1    = (const float*)d_in[0];
  const float* xyz2    = (const float*)d_in[1];
  const float* points1 = (const float*)d_in[2];
  const float* points2 = (const float*)d_in[3];
  const float* W1      = (const float*)d_in[4];
  const float* b1      = (const float*)d_in[5];
  const float* g1      = (const float*)d_in[6];
  const float* beta1   = (const float*)d_in[7];
  const float* W2      = (const float*)d_in[8];
  const float* b2      = (const float*)d_in[9];
  const float* g2      = (const float*)d_in[10];
  const float* beta2   = (const float*)d_in[11];

  char* ws = (char*)d_ws;
  // workspace layout (16B aligned)
  size_t off = 0;
  int*   wsIdx = (int*)  (ws + off); off += (size_t)MTOT * 3 * 4;       // 786,432
  float* wsW   = (float*)(ws + off); off += (size_t)MTOT * 3 * 4;       // 786,432
  float* wsX   = (float*)(ws + off); off += (size_t)MTOT * CIN * 4;     // 100.7 MB
  float* wsY1  = (float*)(ws + off); off += (size_t)MTOT * O1 * 4;      // 67 MB
  float* st    = (float*)(ws + off);                                    // stats region
  float* s1sum = st;           float* s1sq  = st + 256;
  float* s2sum = st + 512;     float* s2sq  = st + 640;
  float* ss1sc = st + 768;     float* ss1sh = st + 1024;
  float* ss2sc = st + 1280;    float* ss2sh = st + 1408;
  float* outF  = (float*)d_out;

  // 1) KNN
  knn_kernel<<<dim3(NN / 256, BB), 256, 0, stream>>>(xyz1, xyz2, wsIdx, wsW);

  // 2) interpolate + concat -> X
  {
    const size_t total = (size_t)MTOT * (CIN / 4);
    interp_concat_kernel<<<(unsigned)(total / 256), 256, 0, stream>>>(
        points1, points2, wsIdx, wsW, wsX);
  }

  // 3) zero BN stats accumulators (768 floats)
  zero_stats_kernel<<<3, 256, 0, stream>>>(st, 768);

  // 4) GEMM1 (fp32 WMMA) + bias + BN stats
  gemm_wmma_kernel<CIN><<<dim3(MTOT / 128, O1 / 32), 256, 0, stream>>>(
      wsX, W1, b1, nullptr, nullptr, wsY1, s1sum, s1sq, O1);

  // 5) finalize BN1
  finalize_stats_kernel<<<1, 256, 0, stream>>>(s1sum, s1sq, g1, beta1, ss1sc, ss1sh, O1);

  // 6) GEMM2 with fused BN1+ReLU on input; raw output into d_out; BN2 stats
  gemm_wmma_kernel<O1><<<dim3(MTOT / 128, O2 / 32), 256, 0, stream>>>(
      wsY1, W2, b2, ss1sc, ss1sh, outF, s2sum, s2sq, O2);

  // 7) finalize BN2
  finalize_stats_kernel<<<1, 256, 0, stream>>>(s2sum, s2sq, g2, beta2, ss2sc, ss2sh, O2);

  // 8) BN2 + ReLU in place
  bnrelu_out_kernel<<<(unsigned)((size_t)MTOT * O2 / 4 / 256), 256, 0, stream>>>(
      outF, ss2sc, ss2sh);
}